// v2loss_33260226740226
// MI455X (gfx1250) — compile-verified
//
#include <hip/hip_runtime.h>
#include <hip/hip_bf16.h>

// ---- problem constants (match reference) ----
#define BS_    2048
#define H_     14
#define W_     14
#define A_     5
#define NC_    2
#define G_     30
#define SZ_    7                      // 1 + NC + 4
#define CELLS_ (H_*W_)                // 196
#define NBOX_  (CELLS_*A_)            // 980
#define PREDF_ (A_*SZ_*CELLS_)        // 6860 floats per image
#define LABF_  (7*CELLS_)             // 1372 floats per image

__device__ __forceinline__ float sigm_(float x) { return 1.0f / (1.0f + __expf(-x)); }

// Low 32 bits of a generic pointer into __shared__ are the wave-relative LDS
// byte offset (generic = {SHARED_BASE[hi32], lds_offset[31:0]}).
__device__ __forceinline__ unsigned lds_off_of_(const void* p) {
  return (unsigned)(unsigned long long)p;
}

// CDNA5 async global->LDS copy, GVS addressing: saddr(base) + vaddr(off).
// VDST = LDS byte address VGPR. Tracked by ASYNCcnt.
__device__ __forceinline__ void async_ld_b128_(unsigned lds_byte_off,
                                               unsigned g_byte_off,
                                               const void* sbase) {
  asm volatile("global_load_async_to_lds_b128 %0, %1, %2"
               :
               : "v"(lds_byte_off), "v"(g_byte_off), "s"(sbase)
               : "memory");
}

__device__ __forceinline__ void wait_async0_() {
  asm volatile("s_wait_asynccnt 0" ::: "memory");
}

__global__ __launch_bounds__(256) void yolo2_loss_kernel(
    const float* __restrict__ pred,     // (BS, 35, 14, 14)
    const float* __restrict__ label,    // (BS, 7, 14, 14)
    const float* __restrict__ anchors,  // (5, 2)
    const int*   __restrict__ seen_p,   // scalar
    float*       __restrict__ out)      // (1,)
{
  __shared__ __align__(16) float s_pred[PREDF_];   // 27440 B
  __shared__ __align__(16) float s_lab[LABF_];     //  5488 B
  __shared__ float s_m[NBOX_];                     //  3920 B
  __shared__ float s_ta[NBOX_ * 5];                // 19600 B
  __shared__ float s_gx[G_], s_gy[G_], s_tw[G_], s_th[G_];
  __shared__ int   s_cell[G_];
  __shared__ int   s_nv;
  __shared__ float s_sum;

  const int tid = threadIdx.x;
  const int b   = blockIdx.x;
  const float* pbase = pred  + (size_t)b * PREDF_;
  const float* lbase = label + (size_t)b * LABF_;

  // ---- Phase 0: async-stage this image's pred+label slices into LDS ----
  {
    const int NP = PREDF_ / 4;                       // 1715 b128 chunks
    for (int i = tid; i < NP; i += 256)
      async_ld_b128_(lds_off_of_(&s_pred[i * 4]), (unsigned)(i * 16), pbase);
    const int NL = LABF_ / 4;                        // 343 b128 chunks
    for (int i = tid; i < NL; i += 256)
      async_ld_b128_(lds_off_of_(&s_lab[i * 4]), (unsigned)(i * 16), lbase);
  }
  // zero scatter accumulators while async copies are in flight
  for (int i = tid; i < NBOX_;     i += 256) s_m[i]  = 0.0f;
  for (int i = tid; i < NBOX_ * 5; i += 256) s_ta[i] = 0.0f;
  if (tid == 0) { s_sum = 0.0f; s_nv = 0; }
  wait_async0_();
  __syncthreads();

  // ---- Phase 1: ordered scan of objectness map (== top_k on binary map) ----
  if (tid == 0) {
    int n = 0;
    for (int cell = 0; cell < CELLS_ && n < G_; ++cell)
      if (s_lab[cell] > 0.0f) s_cell[n++] = cell;
    s_nv = n;
  }
  __syncthreads();
  const int nv = s_nv;

  // anchor_bias = anchors / (512/14) / [W,H]   (same float op order as ref)
  float ab_x[A_], ab_y[A_];
#pragma unroll
  for (int k = 0; k < A_; ++k) {
    ab_x[k] = anchors[2 * k + 0] / (512.0f / 14.0f) / (float)W_;
    ab_y[k] = anchors[2 * k + 1] / (512.0f / 14.0f) / (float)H_;
  }

  float acc = 0.0f;

  // ---- Phase 2: per-truth params + scatter-add (m, ta) + class loss ----
  if (tid < nv) {
    const int cell = s_cell[tid];
    const int r = cell / W_, c = cell % W_;
    const float tx = s_lab[3 * CELLS_ + cell];
    const float ty = s_lab[4 * CELLS_ + cell];
    const float tw = s_lab[5 * CELLS_ + cell];
    const float th = s_lab[6 * CELLS_ + cell];
    const float gx = (tx + (float)c) / (float)W_;
    const float gy = (ty + (float)r) / (float)H_;
    s_gx[tid] = gx; s_gy[tid] = gy; s_tw[tid] = tw; s_th[tid] = th;

    // anchor assignment: argmax IOU of (tw,th) vs anchor at origin (first-max)
    int aid = 0; float best = -1.0f;
#pragma unroll
    for (int k = 0; k < A_; ++k) {
      float ia = fminf(tw, ab_x[k]) * fminf(th, ab_y[k]);
      float ratio = ia / (tw * th + ab_x[k] * ab_y[k] - ia);
      if (ratio > best) { best = ratio; aid = k; }
    }

    const int   wi = (int)(gx * (float)W_);
    const float fx = gx * (float)W_ - (float)wi;
    const int   hj = (int)(gy * (float)H_);
    const float fy = gy * (float)H_ - (float)hj;
    const int pc_cell = hj * W_ + wi;

    const float p0 = s_pred[(aid * SZ_ + 0) * CELLS_ + pc_cell];
    const float p1 = s_pred[(aid * SZ_ + 1) * CELLS_ + pc_cell];
    const float p2 = s_pred[(aid * SZ_ + 2) * CELLS_ + pc_cell];
    const float px = s_pred[(aid * SZ_ + 3) * CELLS_ + pc_cell];
    const float py = s_pred[(aid * SZ_ + 4) * CELLS_ + pc_cell];
    const float pw = s_pred[(aid * SZ_ + 5) * CELLS_ + pc_cell];
    const float ph = s_pred[(aid * SZ_ + 6) * CELLS_ + pc_cell];
    const float sx = sigm_(px), sy = sigm_(py);

    // iou_sel: pred box at (hj,wi,aid) vs this truth
    const float x1 = (sx + (float)wi) / (float)W_;
    const float y1 = (sy + (float)hj) / (float)H_;
    const float w1 = __expf(pw) * ab_x[aid];
    const float h1 = __expf(ph) * ab_y[aid];
    const float xi1 = fmaxf(x1 - 0.5f * w1, gx - 0.5f * tw);
    const float yi1 = fmaxf(y1 - 0.5f * h1, gy - 0.5f * th);
    const float xi2 = fminf(x1 + 0.5f * w1, gx + 0.5f * tw);
    const float yi2 = fminf(y1 + 0.5f * h1, gy + 0.5f * th);
    const float inter = fmaxf(xi2 - xi1, 0.0f) * fmaxf(yi2 - yi1, 0.0f);
    const float iou = fmaxf(inter / (w1 * h1 + tw * th - inter), 0.0f);

    const float wgt = 1.0f * (2.0f - fx * fy);   // COORD_SCALE
    const float t0 = wgt * (sx - fx);
    const float t1 = wgt * (sy - fy);
    const float t2 = wgt * (pw - __logf(tw / ab_x[aid]));
    const float t3 = wgt * (ph - __logf(th / ab_y[aid]));
    const float t4 = 5.0f * (p0 - iou);          // OBJ_SCALE

    const int bidx = pc_cell * A_ + aid;
    atomicAdd(&s_m[bidx], 1.0f);
    atomicAdd(&s_ta[bidx * 5 + 0], t0);
    atomicAdd(&s_ta[bidx * 5 + 1], t1);
    atomicAdd(&s_ta[bidx * 5 + 2], t2);
    atomicAdd(&s_ta[bidx * 5 + 3], t3);
    atomicAdd(&s_ta[bidx * 5 + 4], t4);

    // class loss (CLASS_SCALE == 1)
    const float d1 = p1 - s_lab[1 * CELLS_ + pc_cell];
    const float d2 = p2 - s_lab[2 * CELLS_ + pc_cell];
    acc += d1 * d1 + d2 * d2;
  }
  __syncthreads();

  // ---- Phase 3: per-box no-obj/prior + combine with scattered targets ----
  const bool prior_on = (seen_p[0] < 12800);
  for (int idx = tid; idx < NBOX_; idx += 256) {
    const int cell = idx / A_;
    const int a    = idx % A_;
    const int r = cell / W_, c = cell % W_;

    const float p0 = s_pred[(a * SZ_ + 0) * CELLS_ + cell];
    const float px = s_pred[(a * SZ_ + 3) * CELLS_ + cell];
    const float py = s_pred[(a * SZ_ + 4) * CELLS_ + cell];
    const float pw = s_pred[(a * SZ_ + 5) * CELLS_ + cell];
    const float ph = s_pred[(a * SZ_ + 6) * CELLS_ + cell];
    const float sx = sigm_(px), sy = sigm_(py);
    const float x1 = (sx + (float)c) / (float)W_;
    const float y1 = (sy + (float)r) / (float)H_;
    const float w1 = __expf(pw) * ab_x[a];
    const float h1 = __expf(ph) * ab_y[a];

    float maxiou = 0.0f;  // invalid truth slots contribute iou==0 in ref
    for (int g = 0; g < nv; ++g) {
      const float gx = s_gx[g], gy = s_gy[g], tw = s_tw[g], th = s_th[g];
      const float xi1 = fmaxf(x1 - 0.5f * w1, gx - 0.5f * tw);
      const float yi1 = fmaxf(y1 - 0.5f * h1, gy - 0.5f * th);
      const float xi2 = fminf(x1 + 0.5f * w1, gx + 0.5f * tw);
      const float yi2 = fminf(y1 + 0.5f * h1, gy + 0.5f * th);
      const float inter = fmaxf(xi2 - xi1, 0.0f) * fmaxf(yi2 - yi1, 0.0f);
      const float iou = fmaxf(inter / (w1 * h1 + tw * th - inter), 0.0f);
      maxiou = fmaxf(maxiou, iou);
    }

    float co0, co1, co2, co3;
    if (prior_on) {
      co0 = 0.01f * (sx - 0.5f);
      co1 = 0.01f * (sy - 0.5f);
      co2 = 0.01f * pw;
      co3 = 0.01f * ph;
    } else {
      co0 = co1 = co2 = co3 = 0.0f;
    }
    const float co4 = (maxiou <= 0.6f) ? 0.5f * p0 : 0.0f;  // NOOBJ_SCALE

    const float m  = s_m[idx];
    const float im = 1.0f - m;
    float v;
    v = m * s_ta[idx * 5 + 0] + im * co0; acc += v * v;
    v = m * s_ta[idx * 5 + 1] + im * co1; acc += v * v;
    v = m * s_ta[idx * 5 + 2] + im * co2; acc += v * v;
    v = m * s_ta[idx * 5 + 3] + im * co3; acc += v * v;
    v = m * s_ta[idx * 5 + 4] + im * co4; acc += v * v;
  }

  // ---- Phase 4: block reduction, gate on any(valid), global accumulate ----
#pragma unroll
  for (int off = 16; off > 0; off >>= 1)
    acc += __shfl_down(acc, off, 32);
  if ((tid & 31) == 0) atomicAdd(&s_sum, acc);
  __syncthreads();
  if (tid == 0 && s_nv > 0)
    atomicAdd(out, s_sum * (1.0f / (float)BS_));
}

extern "C" void kernel_launch(void* const* d_in, const int* in_sizes, int n_in,
                              void* d_out, int out_size, void* d_ws, size_t ws_size,
                              hipStream_t stream) {
  (void)in_sizes; (void)n_in; (void)d_ws; (void)ws_size; (void)out_size;
  const float* pred    = (const float*)d_in[0];
  const float* label   = (const float*)d_in[1];
  const float* anchors = (const float*)d_in[2];
  const int*   seen    = (const int*)d_in[3];
  float* out = (float*)d_out;

  hipMemsetAsync(out, 0, sizeof(float), stream);  // graph-capture safe
  yolo2_loss_kernel<<<BS_, 256, 0, stream>>>(pred, label, anchors, seen, out);
}